// FlowSAN_24446953849545
// MI455X (gfx1250) — compile-verified
//
#include <hip/hip_runtime.h>
#include <math.h>

typedef __attribute__((ext_vector_type(2))) float v2f;
typedef __attribute__((ext_vector_type(8))) float v8f;

constexpr int E_    = 8192;
constexpr int NNZN  = 131072;
constexpr int B_    = 64;
constexpr int OUT_  = 10;

// ---------------------------------------------------------------- utilities
__global__ void zero_f(float* __restrict__ p, int n) {
    int i = blockIdx.x * blockDim.x + threadIdx.x;
    if (i < n) p[i] = 0.0f;
}

__device__ inline void atomicMaxF(float* addr, float val) {
    int* ai = reinterpret_cast<int*>(addr);
    int old = __float_as_int(*addr);
    while (__int_as_float(old) < val) {
        int assumed = old;
        old = atomicCAS(ai, assumed, __float_as_int(val));
        if (old == assumed) break;
    }
}

// ------------------------------------------------------- Laplacian diagonal
__global__ void diag_acc(const int* __restrict__ r, const int* __restrict__ c,
                         const float* __restrict__ val, float* __restrict__ d, int nnz) {
    int i = blockIdx.x * blockDim.x + threadIdx.x;
    if (i >= nnz) return;
    int rr = r[i];
    if (rr == c[i]) atomicAdd(&d[rr], val[i]);
}

__global__ void make_dinv(const float* __restrict__ d, float* __restrict__ dinv, int n) {
    int i = blockIdx.x * blockDim.x + threadIdx.x;
    if (i >= n) return;
    float v = d[i];
    dinv[i] = (v > 0.0f) ? 1.0f / sqrtf(fmaxf(v, 1e-12f)) : 0.0f;
}

// -------------------------------------------- pad layer-4 weights 10 -> 16
__global__ void pad_l4(const float* __restrict__ W4d, const float* __restrict__ W4p,
                       const float* __restrict__ a4s, const float* __restrict__ a4d,
                       float* __restrict__ W4dP, float* __restrict__ W4pP,
                       float* __restrict__ a4sP, float* __restrict__ a4dP) {
    int i = blockIdx.x * blockDim.x + threadIdx.x;
    if (i < 32 * 16) {
        int rr = i / 16, cc = i % 16;
        W4dP[i] = (cc < OUT_) ? W4d[rr * OUT_ + cc] : 0.0f;
        W4pP[i] = (cc < OUT_) ? W4p[rr * OUT_ + cc] : 0.0f;
        if (i < 16) {
            a4sP[i] = (i < OUT_) ? a4s[i] : 0.0f;
            a4dP[i] = (i < OUT_) ? a4d[i] : 0.0f;
        }
    }
}

// ---------------------------------------------------- fp32 WMMA tiled GEMM
// C[E_ x N] = A[E_ x K] * B[K x N], one wave per 16x16 tile, K multiple of 4.
__global__ void __launch_bounds__(32)
wmma_gemm_f32(const float* __restrict__ A, const float* __restrict__ Bm,
              float* __restrict__ C, int K, int N) {
    const int ntn  = N >> 4;
    const int tm   = blockIdx.x / ntn;
    const int tn   = blockIdx.x % ntn;
    const int lane = threadIdx.x & 31;
    const int half = lane >> 4;     // 0: K+{0,1}, 1: K+{2,3}
    const int l15  = lane & 15;

    v8f c = {};
    const float* arow = A + (tm * 16 + l15) * K;   // A row for M = l15
    const float* bcol = Bm + tn * 16 + l15;        // B column for N = l15
    for (int k = 0; k < K; k += 4) {
        const int ka = k + 2 * half;
        v2f a, b;
        a.x = arow[ka];
        a.y = arow[ka + 1];
        b.x = bcol[(ka    ) * N];
        b.y = bcol[(ka + 1) * N];
        // (neg_a, A, neg_b, B, c_mod, C, reuse_a, reuse_b)
        c = __builtin_amdgcn_wmma_f32_16x16x4_f32(false, a, false, b, (short)0, c,
                                                  false, false);
    }
    // C/D layout: VGPR j holds row (j + 8*half), col l15
    float* cp = C + (tm * 16 + 8 * half) * N + tn * 16 + l15;
#pragma unroll
    for (int j = 0; j < 8; ++j) cp[j * N] = c[j];
}

// ------------------------------------------------------- sparse L @ Z SpMM
__global__ void spmm_acc(const int* __restrict__ r, const int* __restrict__ c,
                         const float* __restrict__ val, const float* __restrict__ dinv,
                         const float* __restrict__ Z, float* __restrict__ acc,
                         int doP, int nnz) {
    int i = blockIdx.x * blockDim.x + threadIdx.x;
    if (i >= nnz) return;
    int rr = r[i], cc = c[i];
    float w = val[i] * dinv[rr] * dinv[cc];
    const float* zp = Z + cc * doP;
    float* ap = acc + rr * doP;
    for (int f = 0; f < doP; ++f) atomicAdd(&ap[f], w * zp[f]);
}

// ------------------------------------------------------------- GAT kernels
__global__ void row_dots(const float* __restrict__ Hd, const float* __restrict__ a_s,
                         const float* __restrict__ a_d, float* __restrict__ s,
                         float* __restrict__ t, int doP, int n) {
    int i = blockIdx.x * blockDim.x + threadIdx.x;
    if (i >= n) return;
    float ss = 0.0f, tt = 0.0f;
    const float* h = Hd + i * doP;
    for (int f = 0; f < doP; ++f) { ss += h[f] * a_s[f]; tt += h[f] * a_d[f]; }
    s[i] = ss; t[i] = tt;
}

__global__ void init_md(float* __restrict__ m, float* __restrict__ den, int n) {
    int i = blockIdx.x * blockDim.x + threadIdx.x;
    if (i >= n) return;
    m[i] = -3.0e38f;
    den[i] = 0.0f;
}

__global__ void gat_e_max(const int* __restrict__ r, const int* __restrict__ c,
                          const float* __restrict__ s, const float* __restrict__ t,
                          float* __restrict__ e, float* __restrict__ m, int nnz) {
    int i = blockIdx.x * blockDim.x + threadIdx.x;
    if (i >= nnz) return;
    float v = s[r[i]] + t[c[i]];
    v = (v >= 0.0f) ? v : 0.2f * v;     // LeakyReLU(0.2)
    e[i] = v;
    atomicMaxF(&m[r[i]], v);
}

__global__ void gat_exp_den(const int* __restrict__ r, const float* __restrict__ m,
                            float* __restrict__ e, float* __restrict__ den, int nnz) {
    int i = blockIdx.x * blockDim.x + threadIdx.x;
    if (i >= nnz) return;
    float ex = expf(e[i] - m[r[i]]);
    e[i] = ex;
    atomicAdd(&den[r[i]], ex);
}

__global__ void gat_scatter(const int* __restrict__ r, const int* __restrict__ c,
                            const float* __restrict__ e, const float* __restrict__ den,
                            const float* __restrict__ Hd, float* __restrict__ acc,
                            int doP, int nnz) {
    int i = blockIdx.x * blockDim.x + threadIdx.x;
    if (i >= nnz) return;
    int rr = r[i], cc = c[i];
    float alpha = e[i] / (den[rr] + 1e-16f);
    const float* h = Hd + cc * doP;
    float* ap = acc + rr * doP;
    for (int f = 0; f < doP; ++f) atomicAdd(&ap[f], alpha * h[f]);
}

__global__ void relu_copy(const float* __restrict__ acc, float* __restrict__ x, int n) {
    int i = blockIdx.x * blockDim.x + threadIdx.x;
    if (i < n) x[i] = fmaxf(acc[i], 0.0f);
}

// ------------------------------------------------------------ pool+softmax
__global__ void pool_acc(const float* __restrict__ x, const int* __restrict__ batch1,
                         float* __restrict__ sums, float* __restrict__ cnt, int n) {
    int i = blockIdx.x * blockDim.x + threadIdx.x;
    if (i >= n) return;
    int b = batch1[i];
    const float* xp = x + i * 16;   // layer-4 output stride (padded 16)
    for (int f = 0; f < OUT_; ++f) atomicAdd(&sums[b * OUT_ + f], xp[f]);
    atomicAdd(&cnt[b], 1.0f);
}

__global__ void pool_softmax(const float* __restrict__ sums, const float* __restrict__ cnt,
                             float* __restrict__ out) {
    int b = blockIdx.x * blockDim.x + threadIdx.x;
    if (b >= B_) return;
    float cn = fmaxf(cnt[b], 1.0f);
    float p[OUT_];
    float mx = -3.0e38f;
    for (int f = 0; f < OUT_; ++f) { p[f] = sums[b * OUT_ + f] / cn; mx = fmaxf(mx, p[f]); }
    float se = 0.0f;
    for (int f = 0; f < OUT_; ++f) { p[f] = expf(p[f] - mx); se += p[f]; }
    for (int f = 0; f < OUT_; ++f) out[b * OUT_ + f] = p[f] / se;
}

// ------------------------------------------------------------------- driver
extern "C" void kernel_launch(void* const* d_in, const int* in_sizes, int n_in,
                              void* d_out, int out_size, void* d_ws, size_t ws_size,
                              hipStream_t stream) {
    (void)in_sizes; (void)n_in; (void)out_size; (void)ws_size;

    const float* X1     = (const float*)d_in[0];
    const int*   idx_u  = (const int*)  d_in[1];
    const float* val_u  = (const float*)d_in[2];
    const int*   idx_d  = (const int*)  d_in[3];
    const float* val_d  = (const float*)d_in[4];
    const int*   batch1 = (const int*)  d_in[5];
    const float *Wd[4], *As[4], *Ad[4], *Wp[4];
    for (int l = 0; l < 4; ++l) {
        Wd[l] = (const float*)d_in[6 + 4 * l];
        As[l] = (const float*)d_in[7 + 4 * l];
        Ad[l] = (const float*)d_in[8 + 4 * l];
        Wp[l] = (const float*)d_in[9 + 4 * l];
    }

    float* ws   = (float*)d_ws;
    float* dinv = ws; ws += E_;
    float* dwk  = ws; ws += E_;
    float* s    = ws; ws += E_;
    float* t    = ws; ws += E_;
    float* m    = ws; ws += E_;
    float* den  = ws; ws += E_;
    float* Hd   = ws; ws += E_ * 32;
    float* Z    = ws; ws += E_ * 32;
    float* acc  = ws; ws += E_ * 32;
    float* xA   = ws; ws += E_ * 32;
    float* xB   = ws; ws += E_ * 32;
    float* esc  = ws; ws += NNZN;
    float* W4dP = ws; ws += 32 * 16;
    float* W4pP = ws; ws += 32 * 16;
    float* a4sP = ws; ws += 16;
    float* a4dP = ws; ws += 16;
    float* sums = ws; ws += B_ * OUT_;
    float* cnt  = ws; ws += B_;

    const int TB = 256;
    const int gE   = (E_ + TB - 1) / TB;
    const int gNNZ = (NNZN + TB - 1) / TB;

    // diagonal -> dinv
    zero_f<<<gE, TB, 0, stream>>>(dwk, E_);
    diag_acc<<<gNNZ, TB, 0, stream>>>(idx_u, idx_u + NNZN, val_u, dwk, NNZN);
    diag_acc<<<gNNZ, TB, 0, stream>>>(idx_d, idx_d + NNZN, val_d, dwk, NNZN);
    make_dinv<<<gE, TB, 0, stream>>>(dwk, dinv, E_);

    // pad layer-4 params to 16 columns
    pad_l4<<<2, TB, 0, stream>>>(Wd[3], Wp[3], As[3], Ad[3], W4dP, W4pP, a4sP, a4dP);

    auto layer = [&](const float* x, int di, const float* Wd_, const float* as_,
                     const float* ad_, const float* Wp_, int doP, float* xout) {
        const int tiles = (E_ / 16) * (doP / 16);
        wmma_gemm_f32<<<tiles, 32, 0, stream>>>(x, Wd_, Hd, di, doP);
        wmma_gemm_f32<<<tiles, 32, 0, stream>>>(x, Wp_, Z, di, doP);

        const int gA = (E_ * doP + TB - 1) / TB;
        zero_f<<<gA, TB, 0, stream>>>(acc, E_ * doP);

        // h_p = Lnorm @ Z  (sparse, both halves)
        spmm_acc<<<gNNZ, TB, 0, stream>>>(idx_u, idx_u + NNZN, val_u, dinv, Z, acc, doP, NNZN);
        spmm_acc<<<gNNZ, TB, 0, stream>>>(idx_d, idx_d + NNZN, val_d, dinv, Z, acc, doP, NNZN);

        // GAT attention scalars (shared by both passes)
        row_dots<<<gE, TB, 0, stream>>>(Hd, as_, ad_, s, t, doP, E_);

        for (int p = 0; p < 2; ++p) {
            const int* ri = (p == 0) ? idx_u : idx_d;
            init_md<<<gE, TB, 0, stream>>>(m, den, E_);
            gat_e_max  <<<gNNZ, TB, 0, stream>>>(ri, ri + NNZN, s, t, esc, m, NNZN);
            gat_exp_den<<<gNNZ, TB, 0, stream>>>(ri, m, esc, den, NNZN);
            gat_scatter<<<gNNZ, TB, 0, stream>>>(ri, ri + NNZN, esc, den, Hd, acc, doP, NNZN);
        }
        relu_copy<<<gA, TB, 0, stream>>>(acc, xout, E_ * doP);
    };

    layer(X1, 64, Wd[0], As[0], Ad[0], Wp[0], 32, xA);
    layer(xA, 32, Wd[1], As[1], Ad[1], Wp[1], 32, xB);
    layer(xB, 32, Wd[2], As[2], Ad[2], Wp[2], 32, xA);
    layer(xA, 32, W4dP,  a4sP,  a4dP,  W4pP,  16, xB);

    // mean pool per graph + softmax
    zero_f<<<(B_ * OUT_ + TB - 1) / TB, TB, 0, stream>>>(sums, B_ * OUT_);
    zero_f<<<1, TB, 0, stream>>>(cnt, B_);
    pool_acc<<<gE, TB, 0, stream>>>(xB, batch1, sums, cnt, E_);
    pool_softmax<<<1, B_, 0, stream>>>(sums, cnt, (float*)d_out);
}